// EqualizedFocalLoss1_54417235640835
// MI455X (gfx1250) — compile-verified
//
#include <hip/hip_runtime.h>
#include <hip/hip_bf16.h>

// ---------------- problem constants ----------------
constexpr int kB    = 32;
constexpr int kNcls = 15;
constexpr int kH    = 256;
constexpr int kW    = 256;
constexpr int kK    = 500;
constexpr int kCreg = 2;
constexpr int kHW   = kH * kW;            // 65536
constexpr int kNPts = kB * kK;            // 16000
constexpr int kPlanes = kB * kNcls;       // 480

constexpr float kLn2 = 0.69314718055994530942f;

__constant__ float c_gammas[kNcls] = {2.7f,2.1f,2.4f,2.0f,3.0f,2.9f,3.0f,2.5f,
                                      2.1f,2.6f,2.0f,2.1f,2.7f,2.4f,2.2f};

typedef __attribute__((ext_vector_type(2))) float v2f;
typedef __attribute__((ext_vector_type(8))) float v8f;

// ---- raw hardware transcendentals: v_log_f32 (log2) / v_exp_f32 (exp2) ----
__device__ __forceinline__ float fast_log2(float x) {
#if __has_builtin(__builtin_amdgcn_logf)
  return __builtin_amdgcn_logf(x);
#else
  return __log2f(x);
#endif
}
__device__ __forceinline__ float fast_exp2(float x) {
#if __has_builtin(__builtin_amdgcn_exp2f)
  return __builtin_amdgcn_exp2f(x);
#else
  return exp2f(x);
#endif
}
// ln(x) = ln2 * log2(x);  x^g = exp2(g * log2(x))  (x=0 -> -inf -> exp2 -> 0, correct)

// cross-half (lanes n <-> n+16) add via ds_swizzle SWAPX16 (xor=0x10,and=0x1f -> 0x401F)
__device__ __forceinline__ float xhalf_add(float t) {
#if __has_builtin(__builtin_amdgcn_ds_swizzle)
  int o = __builtin_amdgcn_ds_swizzle(__float_as_int(t), 0x401F);
  return t + __int_as_float(o);
#else
  return t + __shfl_xor(t, 16, 32);
#endif
}

// Per-element focal contribution (pos term when g==1, neg term when g<1)
__device__ __forceinline__ float fcontrib(float p, float g, float gamma) {
  const float eps = 1e-12f;
  float np = 1.0f - p;
  float r = 0.0f;
  if (g == 1.0f) {
    r += (kLn2 * fast_log2(p + eps)) * fast_exp2(gamma * fast_log2(np));
  }
  if (g < 1.0f) {
    float ng = 1.0f - g;
    float nw = ng * ng; nw *= nw;                // (1-g)^4
    r += (kLn2 * fast_log2(np + eps)) * fast_exp2(gamma * fast_log2(p)) * nw;
  }
  return r;
}

// ---------------- ws init (ws is poisoned; reset every call) ----------------
__global__ void init_ws_kernel(float* __restrict__ S, float* __restrict__ numpos,
                               int* __restrict__ last) {
  int t = threadIdx.x;
  if (t < kNcls)      S[t] = 0.0f;
  if (t == kNcls)     *numpos = 0.0f;
  if (t == kNcls + 1) *last = -1;
}

// ---------------- per-point huber / arctan factors ----------------
__global__ void points_kernel(const float* __restrict__ output,
                              const int*   __restrict__ mask,
                              const int*   __restrict__ ind,
                              const float* __restrict__ target,
                              float* __restrict__ lbk,
                              float* __restrict__ vals,
                              int*   __restrict__ last) {
  int tid = blockIdx.x * blockDim.x + threadIdx.x;
  if (tid >= kNPts) return;
  int b = tid / kK;
  int p = ind[tid];                                    // 0..HW-1
  float h = 0.0f;
#pragma unroll
  for (int c = 0; c < kCreg; ++c) {
    float pre = output[((size_t)b * kCreg + c) * kHW + p];   // feat[b, p, c]
    float d   = pre - target[(size_t)tid * kCreg + c];
    float ad  = fabsf(d);
    h += (ad < 1.0f) ? 0.5f * d * d : ad - 0.5f;
  }
  float l = h * (1.0f / kCreg);
  lbk[tid] = l;
  float factor = atanf(l) * 0.63661977236758134308f;   // 2/pi
  bool pos = (mask[tid] != 0);
  vals[tid] = pos ? factor : 1.0f;
  if (pos) atomicMax(last, tid);
}

// ---------------- sparse corrections: f(p * prod(vals)) - f(p) ----------------
// One block per batch; exact handling of duplicate (ch,yy,xx) locations
// (inde components are in [0,15) -> heavy duplication, multiplicative compose).
__global__ __launch_bounds__(512) void corr_kernel(const float* __restrict__ pred,
                                                   const float* __restrict__ gt,
                                                   const int*   __restrict__ inde,
                                                   const float* __restrict__ vals,
                                                   float* __restrict__ S) {
  __shared__ int   sloc[kK];
  __shared__ float sval[kK];
  int b = blockIdx.x;
  for (int k = threadIdx.x; k < kK; k += blockDim.x) {
    size_t base = ((size_t)b * kK + k) * 3;
    int ch = inde[base + 0];
    int yy = inde[base + 1];
    int xx = inde[base + 2];
    sloc[k] = (ch << 16) | (yy << 8) | xx;
    sval[k] = vals[b * kK + k];
  }
  __syncthreads();
  int k = threadIdx.x;
  if (k < kK) {
    int myloc = sloc[k];
    float m = 1.0f;
    int first = k;
    for (int j = 0; j < kK; ++j) {
      if (sloc[j] == myloc) {
        m *= sval[j];
        if (j < first) first = j;
      }
    }
    if (first == k && m != 1.0f) {   // first occurrence applies the combined multiplier
      int ch = (myloc >> 16) & 0xFF, yy = (myloc >> 8) & 0xFF, xx = myloc & 0xFF;
      size_t idx = (((size_t)b * kNcls + ch) * kH + yy) * kW + xx;
      float p = pred[idx];
      float g = gt[idx];
      float gamma = c_gammas[ch];
      float delta = fcontrib(p * m, g, gamma) - fcontrib(p, g, gamma);
      atomicAdd(&S[ch], delta);
    }
  }
}

// ---------------- main focal reduction: one 256x256 channel plane per block ----------------
__global__ __launch_bounds__(256) void focal_main_kernel(const float* __restrict__ pred,
                                                         const float* __restrict__ gt,
                                                         float* __restrict__ S,
                                                         float* __restrict__ numpos) {
  const int plane = blockIdx.x;                // b*kNcls + ch
  const int ch    = plane % kNcls;
  const float gamma = c_gammas[ch];            // scalar per block (SGPR)
  const float eps = 1e-12f;

  const float4* p4 = reinterpret_cast<const float4*>(pred) + (size_t)plane * (kHW / 4);
  const float4* g4 = reinterpret_cast<const float4*>(gt)   + (size_t)plane * (kHW / 4);

  float s = 0.0f, cnt = 0.0f;
  constexpr int kIters = kHW / 4 / 256;        // 64 float4 per thread
#pragma unroll 4
  for (int j = 0; j < kIters; ++j) {
    int i = j * 256 + (int)threadIdx.x;
    __builtin_prefetch(p4 + i + 4096, 0, 0);   // global_prefetch_b8 (speculative)
    __builtin_prefetch(g4 + i + 4096, 0, 0);
    float4 pv = p4[i];
    float4 gv = g4[i];
    const float pe[4] = {pv.x, pv.y, pv.z, pv.w};
    const float ge[4] = {gv.x, gv.y, gv.z, gv.w};
#pragma unroll
    for (int e = 0; e < 4; ++e) {
      float p = pe[e], g = ge[e];
      float np = 1.0f - p;
      if (g < 1.0f) {                           // ~always: whole wave takes this path
        float ng = 1.0f - g;
        float nw = ng * ng; nw *= nw;           // (1-g)^4
        // ln(1-p+eps) * p^gamma * (1-g)^4, via 3 hardware trans ops
        s += (kLn2 * fast_log2(np + eps)) * fast_exp2(gamma * fast_log2(p)) * nw;
      } else if (g == 1.0f) {                   // ~never: whole wave skips
        s += (kLn2 * fast_log2(p + eps)) * fast_exp2(gamma * fast_log2(np));
        cnt += 1.0f;
      }
    }
  }

  // ---- cross-lane reduction of (s, cnt) through the matrix pipe ----
  const int lane = threadIdx.x & 31;
  const int wave = threadIdx.x >> 5;
  __shared__ float sh_s[8];
  __shared__ float sh_c[8];

#if __has_builtin(__builtin_amdgcn_wmma_f32_16x16x4_f32)
  // A (16x4 f32): VGPR0 = s (K=0 lanes0-15, K=2 lanes16-31), VGPR1 = cnt (K=1/K=3).
  // B (4x16): rows {0,2} feed columns 0-7 (sum of s), rows {1,3} feed columns 8-15 (sum of cnt).
  v2f a;  a[0] = s; a[1] = cnt;
  int n = lane & 15;
  v2f bsel; bsel[0] = (n < 8) ? 1.0f : 0.0f; bsel[1] = (n < 8) ? 0.0f : 1.0f;
  v8f c = {};
  v8f d = __builtin_amdgcn_wmma_f32_16x16x4_f32(false, a, false, bsel, (short)0, c,
                                                false, false);
  float t = d[0] + d[1] + d[2] + d[3] + d[4] + d[5] + d[6] + d[7]; // 8 D-rows per lane
  t = xhalf_add(t);                            // combine M=0..7 half with M=8..15 half
  // now lanes 0-7 (and 16-23): full wave sum of s ; lanes 8-15 (and 24-31): full wave sum of cnt
  if (lane == 0) sh_s[wave] = t;
  if (lane == 8) sh_c[wave] = t;
#else
  for (int off = 16; off > 0; off >>= 1) {
    s   += __shfl_xor(s,   off, 32);
    cnt += __shfl_xor(cnt, off, 32);
  }
  if (lane == 0) { sh_s[wave] = s; sh_c[wave] = cnt; }
#endif

  __syncthreads();
  if (threadIdx.x == 0) {
    float ts = 0.0f, tc = 0.0f;
#pragma unroll
    for (int w = 0; w < 8; ++w) { ts += sh_s[w]; tc += sh_c[w]; }
    atomicAdd(&S[ch], ts);
    atomicAdd(numpos, tc);
  }
}

// ---------------- finalize ----------------
__global__ void finalize_kernel(const float* __restrict__ S,
                                const float* __restrict__ numpos,
                                const int*   __restrict__ last,
                                const float* __restrict__ lbk,
                                float* __restrict__ out) {
  if (threadIdx.x == 0 && blockIdx.x == 0) {
    float acc = 0.0f;
#pragma unroll
    for (int c = 0; c < kNcls; ++c) acc += S[c] * c_gammas[c] * 0.5f;
    int l = *last;
    float loss0 = (l >= 0) ? lbk[l] : 0.0f;
    float loss = loss0 - acc;
    float npos = *numpos;
    out[0] = (npos == 0.0f) ? loss : loss / npos;
  }
}

extern "C" void kernel_launch(void* const* d_in, const int* in_sizes, int n_in,
                              void* d_out, int out_size, void* d_ws, size_t ws_size,
                              hipStream_t stream) {
  const float* pred   = (const float*)d_in[0];
  const float* gt     = (const float*)d_in[1];
  const float* output = (const float*)d_in[2];
  const int*   mask   = (const int*)  d_in[3];
  const int*   ind    = (const int*)  d_in[4];
  const float* target = (const float*)d_in[5];
  const int*   inde   = (const int*)  d_in[6];

  float* ws_f   = (float*)d_ws;
  float* S      = ws_f;            // 15 per-channel sums
  float* numpos = ws_f + 15;       // 1
  int*   last   = (int*)(ws_f + 16);
  float* lbk    = ws_f + 32;       // kNPts
  float* vals   = ws_f + 32 + kNPts;
  float* out    = (float*)d_out;

  hipLaunchKernelGGL(init_ws_kernel, dim3(1), dim3(64), 0, stream, S, numpos, last);
  hipLaunchKernelGGL(points_kernel, dim3((kNPts + 255) / 256), dim3(256), 0, stream,
                     output, mask, ind, target, lbk, vals, last);
  hipLaunchKernelGGL(corr_kernel, dim3(kB), dim3(512), 0, stream,
                     pred, gt, inde, vals, S);
  hipLaunchKernelGGL(focal_main_kernel, dim3(kPlanes), dim3(256), 0, stream,
                     pred, gt, S, numpos);
  hipLaunchKernelGGL(finalize_kernel, dim3(1), dim3(1), 0, stream,
                     S, numpos, last, lbk, out);
}